// MultiTaskLossWrapper_46703474377042
// MI455X (gfx1250) — compile-verified
//
#include <hip/hip_runtime.h>
#include <hip/hip_bf16.h>
#include <stdint.h>

typedef float v2f __attribute__((ext_vector_type(2)));
typedef float v8f __attribute__((ext_vector_type(8)));

#define B_  4
#define N_  1024
#define M_  2048
#define D_  9
#define K_  64
#define PAD 16      // padded vector stride (K padded 9 -> 12 used, rest zero)
#define CHUNKS 8    // row-split factor for the fused top-k kernel
#define CTILES (N_ / 16)

// workspace layout (in floats)
#define WS_STATS 0                       // [0]=S, [1..9]=mean
#define WS_CINV  16                      // 81 floats
#define WS_AT    128                     // B*M*PAD
#define WS_QT    (WS_AT + B_*M_*PAD)     // B*M
#define WS_GY    (WS_QT + B_*M_)         // B*N*PAD
#define WS_QY    (WS_GY + B_*N_*PAD)     // B*N
#define WS_AY    (WS_QY + B_*N_)         // B*N*PAD
#define WS_QA    (WS_AY + B_*N_*PAD)     // B*N
#define WS_CSI   (WS_QA + B_*N_)         // 1024 col sums (intra)
#define WS_CSR   (WS_CSI + N_)           // 1024 col sums (inter)
#define WS_PARTI (WS_CSR + N_)           // 64*CHUNKS*16*64 partial top-64 (intra)
#define WS_PARTR (WS_PARTI + CTILES*CHUNKS*16*K_)
// total: WS_PARTR + CTILES*CHUNKS*16*K_  ~= 1.33M floats (~5.3 MB)

// ---------------- pass 1: masked sum -> S, mean ----------------
__global__ __launch_bounds__(256)
void stats1_kernel(const float* __restrict__ targets,
                   const uint8_t* __restrict__ mask,
                   float* __restrict__ ws) {
    __shared__ float red[256];
    const int t = threadIdx.x;
    float acc[10];
#pragma unroll
    for (int e = 0; e < 10; ++e) acc[e] = 0.f;
    for (int i = t; i < B_ * M_; i += 256) {
        float w = mask[i] ? 1.f : 0.f;
        const float* tp = targets + (size_t)i * D_;
        acc[0] += w;
#pragma unroll
        for (int d = 0; d < D_; ++d) acc[1 + d] += w * tp[d];
    }
    for (int e = 0; e < 10; ++e) {
        red[t] = acc[e];
        __syncthreads();
        for (int s = 128; s > 0; s >>= 1) {
            if (t < s) red[t] += red[t + s];
            __syncthreads();
        }
        if (t == 0) ws[WS_STATS + e] = red[0];
        __syncthreads();
    }
    if (t == 0) {
        float S = ws[WS_STATS];
        for (int d = 0; d < D_; ++d) ws[WS_STATS + 1 + d] /= S;
    }
}

// ---------------- pass 2: covariance + 9x9 inverse ----------------
__global__ __launch_bounds__(256)
void stats2_kernel(const float* __restrict__ targets,
                   const uint8_t* __restrict__ mask,
                   float* __restrict__ ws) {
    __shared__ float red[256];
    const int t = threadIdx.x;
    float mean[D_];
#pragma unroll
    for (int d = 0; d < D_; ++d) mean[d] = ws[WS_STATS + 1 + d];

    float acc[45];
#pragma unroll
    for (int e = 0; e < 45; ++e) acc[e] = 0.f;
    for (int i = t; i < B_ * M_; i += 256) {
        if (mask[i]) {
            const float* tp = targets + (size_t)i * D_;
            float c[D_];
#pragma unroll
            for (int d = 0; d < D_; ++d) c[d] = tp[d] - mean[d];
            int idx = 0;
#pragma unroll
            for (int d0 = 0; d0 < D_; ++d0)
#pragma unroll
                for (int d1 = d0; d1 < D_; ++d1)
                    acc[idx++] += c[d0] * c[d1];
        }
    }
    float covU[45];
    for (int e = 0; e < 45; ++e) {
        red[t] = acc[e];
        __syncthreads();
        for (int s = 128; s > 0; s >>= 1) {
            if (t < s) red[t] += red[t + s];
            __syncthreads();
        }
        if (t == 0) covU[e] = red[0];
        __syncthreads();
    }
    if (t == 0) {
        float S = ws[WS_STATS];
        float den = S - 1.0f;
        float A[81], Inv[81];
        int idx = 0;
        for (int d0 = 0; d0 < D_; ++d0)
            for (int d1 = d0; d1 < D_; ++d1) {
                float v = covU[idx++] / den;
                A[d0 * 9 + d1] = v;
                A[d1 * 9 + d0] = v;
            }
        for (int i = 0; i < 81; ++i) Inv[i] = 0.f;
        for (int i = 0; i < 9; ++i) Inv[i * 9 + i] = 1.f;
        // Gauss-Jordan with partial pivoting (cov full rank => pinv == inv)
        for (int col = 0; col < 9; ++col) {
            int p = col;
            float best = fabsf(A[col * 9 + col]);
            for (int r = col + 1; r < 9; ++r) {
                float v = fabsf(A[r * 9 + col]);
                if (v > best) { best = v; p = r; }
            }
            if (p != col) {
                for (int c = 0; c < 9; ++c) {
                    float tmp = A[col * 9 + c]; A[col * 9 + c] = A[p * 9 + c]; A[p * 9 + c] = tmp;
                    tmp = Inv[col * 9 + c]; Inv[col * 9 + c] = Inv[p * 9 + c]; Inv[p * 9 + c] = tmp;
                }
            }
            float piv = A[col * 9 + col];
            if (fabsf(piv) < 1e-30f) piv = (piv < 0.f ? -1e-30f : 1e-30f);
            float inv = 1.f / piv;
            for (int c = 0; c < 9; ++c) { A[col * 9 + c] *= inv; Inv[col * 9 + c] *= inv; }
            for (int r = 0; r < 9; ++r) {
                if (r == col) continue;
                float f = A[r * 9 + col];
                for (int c = 0; c < 9; ++c) {
                    A[r * 9 + c]   -= f * A[col * 9 + c];
                    Inv[r * 9 + c] -= f * Inv[col * 9 + c];
                }
            }
        }
        for (int i = 0; i < 81; ++i) ws[WS_CINV + i] = Inv[i];
    }
}

// ---------------- build transformed vectors ----------------
__global__ __launch_bounds__(256)
void build_kernel(const float* __restrict__ outputs,
                  const float* __restrict__ targets,
                  const uint8_t* __restrict__ mask,
                  float* __restrict__ ws) {
    __shared__ float sc[96]; // [0..80]=cinv, [81..89]=mean
    const int tid = threadIdx.x;
    if (tid < 81) sc[tid] = ws[WS_CINV + tid];
    else if (tid < 90) sc[tid] = ws[WS_STATS + 1 + (tid - 81)];
    __syncthreads();
    const float* cinv = sc;
    const float* mean = sc + 81;

    int i = blockIdx.x * 256 + tid;
    if (i < B_ * M_) {
        const float* tp = targets + (size_t)i * D_;
        float a[D_];
#pragma unroll
        for (int d = 0; d < D_; ++d) a[d] = tp[d] - mean[d];
        float q = 0.f;
#pragma unroll
        for (int d = 0; d < D_; ++d) {
            float s = 0.f;
#pragma unroll
            for (int e = 0; e < D_; ++e) s += cinv[d * 9 + e] * a[e];
            q += a[d] * s;
        }
        float* dst = ws + WS_AT + (size_t)i * PAD;
#pragma unroll
        for (int d = 0; d < D_; ++d) dst[d] = a[d];
#pragma unroll
        for (int d = D_; d < PAD; ++d) dst[d] = 0.f;
        ws[WS_QT + i] = mask[i] ? q : __builtin_inff();
    } else if (i < B_ * M_ + B_ * N_) {
        int j = i - B_ * M_;
        const float* yp = outputs + (size_t)j * D_;
        float y[D_], g[D_], a[D_];
#pragma unroll
        for (int d = 0; d < D_; ++d) { y[d] = yp[d]; a[d] = yp[d] - mean[d]; }
        float qy = 0.f, qa = 0.f;
#pragma unroll
        for (int d = 0; d < D_; ++d) {
            float s = 0.f, s2 = 0.f;
#pragma unroll
            for (int e = 0; e < D_; ++e) {
                s  += cinv[d * 9 + e] * y[e];
                s2 += cinv[d * 9 + e] * a[e];
            }
            g[d] = s;  qy += y[d] * s;
            qa += a[d] * s2;
        }
        float* dg = ws + WS_GY + (size_t)j * PAD;
        float* da = ws + WS_AY + (size_t)j * PAD;
#pragma unroll
        for (int d = 0; d < D_; ++d) { dg[d] = g[d]; da[d] = a[d]; }
#pragma unroll
        for (int d = D_; d < PAD; ++d) { dg[d] = 0.f; da[d] = 0.f; }
        ws[WS_QY + j] = qy;
        ws[WS_QA + j] = qa;
    }
}

// ---------------- fused WMMA distance + per-column partial top-64 ----------------
// dist(row,col) = qr[row] + qc[col] - 2 * dot12(R[row], C[col])
// Grid: (64 column tiles) x (CHUNKS row chunks); one wave per block.
// Each block streams its row-tile range through v_wmma_f32_16x16x4_f32
// (3 chained k-steps = K 12); 16 heap-owner lanes keep exact 64-smallest
// max-heaps in LDS; emits partial top-64 lists per column.
__global__ __launch_bounds__(32)
void maha_topk_kernel(const float* __restrict__ R,   // rows*PAD
                      const float* __restrict__ qr,  // rows (inf = masked)
                      const float* __restrict__ Cv,  // B*N*PAD column vectors
                      const float* __restrict__ qc,  // B*N
                      int tilesPerChunk, int rowTilesPerBatch,
                      float* __restrict__ partial) { // [CTILES][CHUNKS][16][64]
    __shared__ float sCv[16 * 16];
    __shared__ float sQc[16];
    __shared__ float sQr[16];
    __shared__ float sTile[16 * 17];
    __shared__ float heap[16 * 65];

    const int lane = threadIdx.x;
    const int ct = blockIdx.x;
    const int chunk = blockIdx.y;
    const int colBase = ct * 16;
    const int rtBegin = chunk * tilesPerChunk;
    const int rtEnd = rtBegin + tilesPerChunk;
    const int m = lane & 15;          // A-row / B-col index for this lane
    const int half = lane >> 4;       // K-half select per ISA 16x4 f32 layout
    const int ks = half * 2;

    for (int i = lane; i < 16 * 65; i += 32) heap[i] = __builtin_inff();
    __syncthreads();
    float rootCache = __builtin_inff();

    int curB = -1;
    for (int rt = rtBegin; rt < rtEnd; ++rt) {
        const int b = rt / rowTilesPerBatch;
        if (b != curB) {
            curB = b;
            const float* src = Cv + ((size_t)b * N_ + colBase) * PAD;
            for (int i = lane; i < 256; i += 32) sCv[i] = src[i];
            if (lane < 16) sQc[lane] = qc[b * N_ + colBase + lane];
        }
        const int rowBase = rt * 16;
        if (lane < 16) sQr[lane] = qr[rowBase + lane];
        __syncthreads();

        const float* rp = R + (size_t)(rowBase + m) * PAD;
        v8f acc = {};
#pragma unroll
        for (int s = 0; s < 3; ++s) {
            v2f a, bb;
            a.x = rp[4 * s + ks];
            a.y = rp[4 * s + ks + 1];
            bb.x = sCv[m * 16 + 4 * s + ks];
            bb.y = sCv[m * 16 + 4 * s + ks + 1];
            acc = __builtin_amdgcn_wmma_f32_16x16x4_f32(
                false, a, false, bb, (short)0, acc, false, false);
        }
        const float qcn = sQc[m];
#pragma unroll
        for (int i = 0; i < 8; ++i) {
            const int rl = i + half * 8;
            sTile[rl * 17 + m] = sQr[rl] + qcn - 2.0f * acc[i];
        }
        __syncthreads();

        if (lane < 16) {
            float* h = &heap[lane * 65];
            for (int j = 0; j < 16; ++j) {
                float v = sTile[j * 17 + lane];
                if (v < rootCache) {
                    int pos = 0;
                    for (;;) {
                        int c = 2 * pos + 1;
                        if (c >= K_) break;
                        float cv = h[c];
                        if (c + 1 < K_) {
                            float cv1 = h[c + 1];
                            if (cv1 > cv) { cv = cv1; ++c; }
                        }
                        if (cv > v) { h[pos] = cv; pos = c; } else break;
                    }
                    h[pos] = v;
                    rootCache = h[0];
                }
            }
        }
        __syncthreads();
    }

    if (lane < 16) {
        const float* h = &heap[lane * 65];
        float* dst = partial + (((size_t)ct * CHUNKS + chunk) * 16 + lane) * K_;
        for (int j = 0; j < K_; ++j) dst[j] = h[j];
    }
}

// ---------------- merge partial top-64 lists, emit per-column sums ----------------
__global__ __launch_bounds__(32)
void topk_merge_kernel(const float* __restrict__ partial, // [CTILES][CHUNKS][16][64]
                       float* __restrict__ colsum) {      // N
    __shared__ float heap[16 * 65];
    const int lane = threadIdx.x;
    const int ct = blockIdx.x;
    for (int i = lane; i < 16 * 65; i += 32) heap[i] = __builtin_inff();
    __syncthreads();

    if (lane < 16) {
        float* h = &heap[lane * 65];
        float root = __builtin_inff();
        for (int ch = 0; ch < CHUNKS; ++ch) {
            const float* p = partial + (((size_t)ct * CHUNKS + ch) * 16 + lane) * K_;
            for (int j = 0; j < K_; ++j) {
                float v = p[j];
                if (v < root) {
                    int pos = 0;
                    for (;;) {
                        int c = 2 * pos + 1;
                        if (c >= K_) break;
                        float cv = h[c];
                        if (c + 1 < K_) {
                            float cv1 = h[c + 1];
                            if (cv1 > cv) { cv = cv1; ++c; }
                        }
                        if (cv > v) { h[pos] = cv; pos = c; } else break;
                    }
                    h[pos] = v;
                    root = h[0];
                }
            }
        }
        float s = 0.f;
        for (int j = 0; j < K_; ++j) s += h[j];
        colsum[ct * 16 + lane] = s;
    }
}

// ---------------- finalize ----------------
__global__ __launch_bounds__(256)
void finalize_kernel(const float* __restrict__ csI,
                     const float* __restrict__ csR,
                     float* __restrict__ out) {
    __shared__ float redA[256];
    __shared__ float redB[256];
    const int t = threadIdx.x;
    float a = 0.f, b = 0.f;
    for (int i = t; i < N_; i += 256) { a += csI[i]; b += csR[i]; }
    redA[t] = a; redB[t] = b;
    __syncthreads();
    for (int s = 128; s > 0; s >>= 1) {
        if (t < s) { redA[t] += redA[t + s]; redB[t] += redB[t + s]; }
        __syncthreads();
    }
    if (t == 0) {
        const float inv = 1.0f / (float)(N_ * K_);
        out[0] = redA[0] * inv - 0.1f * (redB[0] * inv);
    }
}

extern "C" void kernel_launch(void* const* d_in, const int* in_sizes, int n_in,
                              void* d_out, int out_size, void* d_ws, size_t ws_size,
                              hipStream_t stream) {
    (void)in_sizes; (void)n_in; (void)out_size; (void)ws_size;
    const float*   outputs = (const float*)d_in[0];   // (4,1024,9)
    const float*   targets = (const float*)d_in[1];   // (4,2048,9)
    const uint8_t* mask    = (const uint8_t*)d_in[2]; // (4,2048) bool
    float* out = (float*)d_out;
    float* ws  = (float*)d_ws;

    stats1_kernel<<<1, 256, 0, stream>>>(targets, mask, ws);
    stats2_kernel<<<1, 256, 0, stream>>>(targets, mask, ws);
    build_kernel<<<(B_ * M_ + B_ * N_ + 255) / 256, 256, 0, stream>>>(outputs, targets, mask, ws);

    // intra: rows = (b,m) targets (8192 = 512 row tiles), cols = y_pred per batch
    maha_topk_kernel<<<dim3(CTILES, CHUNKS), 32, 0, stream>>>(
        ws + WS_AT, ws + WS_QT, ws + WS_GY, ws + WS_QY,
        (B_ * M_) / 16 / CHUNKS, M_ / 16, ws + WS_PARTI);
    topk_merge_kernel<<<CTILES, 32, 0, stream>>>(ws + WS_PARTI, ws + WS_CSI);

    // inter: rows = (b,n1) y_pred (4096 = 256 row tiles), cols = y_pred - mean per batch
    maha_topk_kernel<<<dim3(CTILES, CHUNKS), 32, 0, stream>>>(
        ws + WS_GY, ws + WS_QY, ws + WS_AY, ws + WS_QA,
        (B_ * N_) / 16 / CHUNKS, N_ / 16, ws + WS_PARTR);
    topk_merge_kernel<<<CTILES, 32, 0, stream>>>(ws + WS_PARTR, ws + WS_CSR);

    finalize_kernel<<<1, 256, 0, stream>>>(ws + WS_CSI, ws + WS_CSR, out);
}